// Model_12910671691946
// MI455X (gfx1250) — compile-verified
//
#include <hip/hip_runtime.h>

typedef __attribute__((ext_vector_type(2))) float v2f;
typedef __attribute__((ext_vector_type(8))) float v8f;

#define NL   4
#define BB   4
#define LL   512
#define DD   768
#define LIN  128
#define ENT  10
#define MTOT (BB*LL)     // 2048
#define NP   32          // padded N (20 -> 32), also S row stride
#define KCH  64          // K chunk per LDS stage
#define TSTR 68          // LDS row stride in floats (16B-aligned rows, odd-ish banking)

// ---------------- Kernel A: fold LIN dimension away ----------------
// U[n][d]: n<10 -> w_cls[n]·w_span[:, :768];  10<=n<20 -> w_cls[n-10]·w_span[:, 768:]; else 0
// konst[c] = b_cls[c] + sum_o b_span[o]*w_cls[c][o]
__global__ void prep_kernel(const float* __restrict__ w_span,
                            const float* __restrict__ b_span,
                            const float* __restrict__ w_cls,
                            const float* __restrict__ b_cls,
                            float* __restrict__ U,
                            float* __restrict__ konst) {
    int e = blockIdx.x * blockDim.x + threadIdx.x;   // 0 .. 32*768-1
    int n = e / DD, d = e % DD;
    float acc = 0.0f;
    if (n < ENT) {
        for (int o = 0; o < LIN; ++o)
            acc += w_cls[n*LIN + o] * w_span[o*(2*DD) + d];
    } else if (n < 2*ENT) {
        int c = n - ENT;
        for (int o = 0; o < LIN; ++o)
            acc += w_cls[c*LIN + o] * w_span[o*(2*DD) + DD + d];
    }
    U[n*DD + d] = acc;
    if (e < ENT) {
        float k = b_cls[e];
        for (int o = 0; o < LIN; ++o) k += b_span[o] * w_cls[e*LIN + o];
        konst[e] = k;
    }
}

// ---------------- Kernel B: S[2048,32] = (sum_layers hs) @ U^T via WMMA f32 16x16x4 ----
// Block = 128 threads = 4 waves. Block owns a 16-row M tile; wave w owns K range [w*192,(w+1)*192).
__global__ void __launch_bounds__(128)
gemm_kernel(const float* __restrict__ hs, const float* __restrict__ U,
            float* __restrict__ S) {
    __shared__ float tokT[4][16*TSTR];   // per-wave A tile (16 x 64, layer-summed)
    __shared__ float uT  [4][32*TSTR];   // per-wave B tile (32 x 64)
    __shared__ float red [4][16*NP];     // per-wave partial C

    const int t    = threadIdx.x;
    const int wq   = t >> 5;
    const int lane = t & 31;
    const int half = lane >> 4;          // K-group for WMMA fragments
    const int r    = lane & 15;
    const int m0   = blockIdx.x * 16;

    v8f c0 = {}; v8f c1 = {};
    const int kbase = wq * 192;

    for (int ch = 0; ch < 3; ++ch) {
        const int kb = kbase + ch * KCH;

        // stage A: 16 rows x 64 cols, summing 4 layers, float4-coalesced
        for (int p = 0; p < 8; ++p) {
            const int row = p*2 + half;            // 0..15
            const int col = r*4;                   // 0..60
            const float* g = hs + (size_t)(m0 + row)*DD + kb + col;
            float4 a0 = *(const float4*)(g);
            float4 a1 = *(const float4*)(g + (size_t)1*MTOT*DD);
            float4 a2 = *(const float4*)(g + (size_t)2*MTOT*DD);
            float4 a3 = *(const float4*)(g + (size_t)3*MTOT*DD);
            float4 s4;
            s4.x = a0.x + a1.x + a2.x + a3.x;
            s4.y = a0.y + a1.y + a2.y + a3.y;
            s4.z = a0.z + a1.z + a2.z + a3.z;
            s4.w = a0.w + a1.w + a2.w + a3.w;
            *(float4*)&tokT[wq][row*TSTR + col] = s4;   // row stride 272B -> 16B aligned
        }
        // stage B: 32 rows x 64 cols of U
        for (int p = 0; p < 16; ++p) {
            const int row = p*2 + half;            // 0..31
            const int col = r*4;
            *(float4*)&uT[wq][row*TSTR + col] = *(const float4*)(U + row*DD + kb + col);
        }
        __syncthreads();

        // 16 K-steps of 4, two N tiles -> 32 WMMAs per chunk per wave
        for (int kk = 0; kk < KCH; kk += 4) {
            // A 16x4: lane r holds {A[r, 2*half], A[r, 2*half+1]}
            v2f a  = *(const v2f*)&tokT[wq][r*TSTR       + kk + 2*half];
            // B 4x16: lane r holds {B[2*half, n0+r], B[2*half+1, n0+r]} = U[n0+r][k..]
            v2f b0 = *(const v2f*)&uT  [wq][r*TSTR       + kk + 2*half];
            v2f b1 = *(const v2f*)&uT  [wq][(16+r)*TSTR  + kk + 2*half];
            c0 = __builtin_amdgcn_wmma_f32_16x16x4_f32(false, a, false, b0, (short)0, c0, false, false);
            c1 = __builtin_amdgcn_wmma_f32_16x16x4_f32(false, a, false, b1, (short)0, c1, false, false);
        }
        __syncthreads();
    }

    // dump per-wave partials: C/D layout VGPR v, lane l -> (M = v + 8*(l/16), N = l%16)
    for (int v = 0; v < 8; ++v) {
        red[wq][(v + 8*half)*NP + r]      = c0[v];
        red[wq][(v + 8*half)*NP + 16 + r] = c1[v];
    }
    __syncthreads();

    // reduce the 4 K-slices and store float4
    const int i4   = t * 4;            // 0..508
    const int mloc = i4 >> 5;
    const int ncol = i4 & 31;
    float4 acc;
    acc.x = red[0][i4+0] + red[1][i4+0] + red[2][i4+0] + red[3][i4+0];
    acc.y = red[0][i4+1] + red[1][i4+1] + red[2][i4+1] + red[3][i4+1];
    acc.z = red[0][i4+2] + red[1][i4+2] + red[2][i4+2] + red[3][i4+2];
    acc.w = red[0][i4+3] + red[1][i4+3] + red[2][i4+3] + red[3][i4+3];
    *(float4*)&S[(size_t)(m0 + mloc)*NP + ncol] = acc;
}

// ---------------- Kernel C: out[b,c,i,j] = S[b*512+i,c] + S[b*512+j,10+c] + konst[c] ----
__global__ void __launch_bounds__(256)
bcast_kernel(const float* __restrict__ S, const float* __restrict__ konst,
             float* __restrict__ out) {
    __shared__ float sj[LL];
    const int t  = threadIdx.x;
    const int bc = blockIdx.y;            // b*10 + c
    const int b  = bc / ENT, c = bc % ENT;
    const int ib = blockIdx.x;            // 8-row strip index

    for (int j = t; j < LL; j += 256)
        sj[j] = S[(size_t)(b*LL + j)*NP + ENT + c];
    __syncthreads();

    const float kc = konst[c];
    const int r2 = t >> 7;                // 0..1
    const int q  = t & 127;
    const int j0 = q * 4;
    const float4 sjv = *(const float4*)&sj[j0];

    for (int rr = 0; rr < 4; ++rr) {
        const int i = ib*8 + rr*2 + r2;
        const float si = S[(size_t)(b*LL + i)*NP + c] + kc;
        float4 o;
        o.x = si + sjv.x; o.y = si + sjv.y; o.z = si + sjv.z; o.w = si + sjv.w;
        *(float4*)&out[((size_t)bc*LL + i)*LL + j0] = o;   // coalesced B128 stores
    }
}

extern "C" void kernel_launch(void* const* d_in, const int* in_sizes, int n_in,
                              void* d_out, int out_size, void* d_ws, size_t ws_size,
                              hipStream_t stream) {
    const float* hs     = (const float*)d_in[0];
    const float* w_span = (const float*)d_in[1];
    const float* b_span = (const float*)d_in[2];
    const float* w_cls  = (const float*)d_in[3];
    const float* b_cls  = (const float*)d_in[4];
    float* out = (float*)d_out;

    // workspace: U[32*768] | konst[32] | S[2048*32]   (~361 KB)
    float* U     = (float*)d_ws;
    float* konst = U + 32*DD;
    float* S     = konst + 32;

    prep_kernel <<<dim3((32*DD)/256), 256, 0, stream>>>(w_span, b_span, w_cls, b_cls, U, konst);
    gemm_kernel <<<dim3(MTOT/16),     128, 0, stream>>>(hs, U, S);
    bcast_kernel<<<dim3(LL/8, BB*ENT), 256, 0, stream>>>(S, konst, out);
}